// GraphAttentionLayer4D_78082505441863
// MI455X (gfx1250) — compile-verified
//
#include <hip/hip_runtime.h>

typedef __attribute__((ext_vector_type(2))) float v2f;
typedef __attribute__((ext_vector_type(8))) float v8f;

#define ALPHA_ 0.2f
#define B_ 8
#define C_ 64
#define T_ 256
#define V_ 128
#define F_ 64

#define WS_STRIDE 65   // ws[c*65 + f]   (pad: rows K and K+2 hit different banks)
#define WH_STRIDE 65   // whs[v*65 + f]  (pad: column access across v conflict-free)

__global__ __launch_bounds__(256)
void gat_diag_kernel(const float* __restrict__ x, const float* __restrict__ W,
                     const float* __restrict__ a, float* __restrict__ out) {
  __shared__ float ws[C_ * WS_STRIDE];     // W[c][f]
  __shared__ float whs[V_ * WH_STRIDE];    // Wh[v][f]
  __shared__ float avec[2 * F_];
  __shared__ float s1s[V_];
  __shared__ float s2s[V_];
  __shared__ float diag[V_];
  __shared__ float red[64];

  const int tid = threadIdx.x;
  const int bt  = blockIdx.x;
  const int b   = bt >> 8;          // T_ == 256
  const int t   = bt & (T_ - 1);

  // ---- stage W and a into LDS (coalesced) ----
  for (int i = tid; i < C_ * F_; i += 256) {
    int c = i >> 6, f = i & 63;
    ws[c * WS_STRIDE + f] = W[i];
  }
  if (tid < 2 * F_) avec[tid] = a[tid];
  __syncthreads();

  // ---- GEMM: Wh[v,f] = sum_c x[b,c,t,v] * W[c,f]  via V_WMMA_F32_16X16X4_F32 ----
  const int wave = tid >> 5;
  const int lane = tid & 31;
  const int half = lane >> 4;       // 0: K,K+1   1: K+2,K+3
  const int l    = lane & 15;
  const int v0   = wave * 16;       // 8 waves cover v = 0..127

  v8f acc0 = {}, acc1 = {}, acc2 = {}, acc3 = {};
  const long xbase = (((long)b * C_) * T_ + t) * (long)V_;   // + c*T_*V_ + v
  const long cstr  = (long)T_ * V_;

  for (int k = 0; k < C_; k += 4) {
    const int c0 = k + 2 * half;
    v2f av;
    av.x = x[xbase + (long)c0 * cstr + v0 + l];
    av.y = x[xbase + (long)(c0 + 1) * cstr + v0 + l];

    v2f bv;
    bv.x = ws[c0 * WS_STRIDE + 0 + l];
    bv.y = ws[(c0 + 1) * WS_STRIDE + 0 + l];
    acc0 = __builtin_amdgcn_wmma_f32_16x16x4_f32(false, av, false, bv, (short)0, acc0, false, false);

    bv.x = ws[c0 * WS_STRIDE + 16 + l];
    bv.y = ws[(c0 + 1) * WS_STRIDE + 16 + l];
    acc1 = __builtin_amdgcn_wmma_f32_16x16x4_f32(false, av, false, bv, (short)0, acc1, false, false);

    bv.x = ws[c0 * WS_STRIDE + 32 + l];
    bv.y = ws[(c0 + 1) * WS_STRIDE + 32 + l];
    acc2 = __builtin_amdgcn_wmma_f32_16x16x4_f32(false, av, false, bv, (short)0, acc2, false, false);

    bv.x = ws[c0 * WS_STRIDE + 48 + l];
    bv.y = ws[(c0 + 1) * WS_STRIDE + 48 + l];
    acc3 = __builtin_amdgcn_wmma_f32_16x16x4_f32(false, av, false, bv, (short)0, acc3, false, false);
  }

  // C/D layout: VGPR r, lanes 0-15 -> M=r, lanes 16-31 -> M=r+8, N=l
#pragma unroll
  for (int r = 0; r < 8; ++r) {
    const int v = v0 + r + 8 * half;
    whs[v * WH_STRIDE +  0 + l] = acc0[r];
    whs[v * WH_STRIDE + 16 + l] = acc1[r];
    whs[v * WH_STRIDE + 32 + l] = acc2[r];
    whs[v * WH_STRIDE + 48 + l] = acc3[r];
  }
  __syncthreads();

  // ---- s1[v] = Wh[v,:].a1 ; s2[v] = Wh[v,:].a2 ----
  {
    const int v = tid & (V_ - 1);
    const float* ap = (tid < V_) ? &avec[0] : &avec[F_];
    float s = 0.f;
#pragma unroll 8
    for (int f = 0; f < F_; ++f) s += whs[v * WH_STRIDE + f] * ap[f];
    if (tid < V_) s1s[v] = s; else s2s[v] = s;
  }
  __syncthreads();

  // ---- max(s2) reduction (leaky-relu is monotone -> row max = L(s1[v] + max s2)) ----
  if (tid < 64) red[tid] = fmaxf(s2s[tid], s2s[tid + 64]);
  __syncthreads();
  if (tid < 32) red[tid] = fmaxf(red[tid], red[tid + 32]);
  __syncthreads();
  if (tid < 16) red[tid] = fmaxf(red[tid], red[tid + 16]);
  __syncthreads();
  if (tid < 8)  red[tid] = fmaxf(red[tid], red[tid + 8]);
  __syncthreads();
  if (tid < 4)  red[tid] = fmaxf(red[tid], red[tid + 4]);
  __syncthreads();
  if (tid < 2)  red[tid] = fmaxf(red[tid], red[tid + 2]);
  __syncthreads();
  if (tid == 0) red[0] = fmaxf(red[0], red[1]);
  __syncthreads();
  const float s2max = red[0];

  // ---- diagonal softmax weight per v ----
  if (tid < V_) {
    const float sv = s1s[tid];
    float mx = sv + s2max;
    mx = mx > 0.f ? mx : ALPHA_ * mx;            // row max of leaky-relu'd logits
    float denom = 0.f;
#pragma unroll 4
    for (int v2 = 0; v2 < V_; ++v2) {
      float e = sv + s2s[v2];
      e = e > 0.f ? e : ALPHA_ * e;
      denom += __expf(e - mx);
    }
    float ed = sv + s2s[tid];
    ed = ed > 0.f ? ed : ALPHA_ * ed;
    diag[tid] = __expf(ed - mx) / denom;
  }
  __syncthreads();

  // ---- out[b,f,t,v] = elu(diag[v] * Wh[v,f])  (v contiguous -> coalesced) ----
  const long obase = (((long)b * F_) * T_ + t) * (long)V_;
  const long fstr  = (long)T_ * V_;
  for (int i = tid; i < F_ * V_; i += 256) {
    const int f = i >> 7;           // V_ == 128
    const int v = i & (V_ - 1);
    const float h = diag[v] * whs[v * WH_STRIDE + f];
    out[obase + (long)f * fstr + v] = h > 0.f ? h : (__expf(h) - 1.f);
  }
}

extern "C" void kernel_launch(void* const* d_in, const int* in_sizes, int n_in,
                              void* d_out, int out_size, void* d_ws, size_t ws_size,
                              hipStream_t stream) {
  (void)in_sizes; (void)n_in; (void)out_size; (void)d_ws; (void)ws_size;
  const float* x = (const float*)d_in[0];
  const float* W = (const float*)d_in[1];
  const float* a = (const float*)d_in[2];
  float* out = (float*)d_out;
  gat_diag_kernel<<<B_ * T_, 256, 0, stream>>>(x, W, a, out);
}